// LocallyConnected2d_11373073399937
// MI455X (gfx1250) — compile-verified
//
#include <hip/hip_runtime.h>

typedef __attribute__((ext_vector_type(2))) float v2f;
typedef __attribute__((ext_vector_type(4))) float v4f;
typedef __attribute__((ext_vector_type(8))) float v8f;

// Problem sizes
#define B_  8
#define C_  32
#define O_  64
#define H_  64
#define W_  64
#define KT_ 9          // 3x3 taps
#define OT  16         // o-tile per block
#define CCH 4          // channels per LDS chunk -> K_chunk = 36 = 9 wmma steps
#define NCHUNK (C_/CCH)

// weight strides (floats): [o][c][h][w][k]
#define WG_O (C_*H_*W_*KT_)   // 1179648
#define WG_C (H_*W_*KT_)      // 36864
#define WG_H (W_*KT_)         // 576

// LDS weight layout: [o(16)][c2(2)][wk(576)][cp(2)]
//  cp innermost -> each B operand (K = v + 2*half) is one aligned b64.
//  S_O ≡ 4, S_C2 ≡ 2 (mod 64) -> wave b64 reads cover 64 distinct banks.
#define WL_C2  1154               // 576*2 + 2 pad
#define WL_O   (2*WL_C2)          // 2308
#define WL_FLOATS (16*WL_O)       // 36928

// LDS x layout: [c2(2)][r(3)][b(16)][q(66)][cp(2)]; q=w+dw; halo q=0,65 and b>=8 stay zero
#define XL_B   132                // 66*2  (≡ 4 mod 64)
#define XL_R   (16*XL_B)          // 2112
#define XL_C2  (3*XL_R + 2)       // 6338  (≡ 2 mod 64)
#define XL_FLOATS (2*XL_C2)       // 12676

#define LDS_FLOATS (WL_FLOATS + XL_FLOATS)   // 49604 floats -> 198416 B < 320 KB

extern "C" __global__ __launch_bounds__(256)
void lc2d_wmma_f32(const float* __restrict__ x,
                   const float* __restrict__ wgt,
                   float* __restrict__ out)
{
    extern __shared__ float lds[];
    float* wl = lds;                 // weights
    float* xl = lds + WL_FLOATS;     // patch source rows

    const int tid  = threadIdx.x;
    const int lane = tid & 31;
    const int wave = tid >> 5;           // 0..7
    const int half = lane >> 4;          // 0|1 -> K-offset +2 in A/B tiles
    const int l15  = lane & 15;

    const int o0 = (blockIdx.x & 3) * OT;    // 0,16,32,48
    const int h  = blockIdx.x >> 2;          // 0..63

    const int wbase = wave * 8;              // this wave's first pixel (w)

    // --- one-time zero of x staging region (halo + batch rows 8..15 persist as zero) ---
    {
        v4f z = v4f{0.f,0.f,0.f,0.f};
        v4f* x4 = (v4f*)xl;
        for (int i = tid; i < XL_FLOATS/4; i += 256) x4[i] = z;
    }

    // Per-lane read bases (dword indices). All inner offsets become immediates.
    // A (patches), 16x4 f32: lane M = l15 (batch; rows 8..15 read zeros), K = v+2*half
    const int aB = half * XL_C2 + l15 * XL_B + wbase * 2;
    // B (weights), 4x16 f32: lane N = o = l15, K = v+2*half ; wk = (wbase+p)*9 + j
    const int bB = l15 * WL_O + half * WL_C2 + wbase * (2*KT_);

    v8f acc[8];
#pragma unroll
    for (int p = 0; p < 8; ++p) acc[p] = v8f{0.f,0.f,0.f,0.f,0.f,0.f,0.f,0.f};

    for (int cc = 0; cc < NCHUNK; ++cc) {
        const int c0 = cc * CCH;
        __syncthreads();   // previous compute (and zero fill) done before overwrite

        // ---- stage weights: thread = one (oo,c2) run, 4 wk (stride 144), both cp ----
        // 8 coalesced scalar global loads + 4 conflict-free interleaving ds_store_b64
        {
            const float* gw = wgt + (size_t)o0 * WG_O + (size_t)c0 * WG_C + (size_t)h * WG_H;
#pragma unroll
            for (int i = 0; i < 18; ++i) {
                const int g   = tid + i * 256;     // [0,4608)
                const int wk0 = g % 144;           // consecutive across lanes
                const int oc  = g / 144;           // [0,32)
                const int c2  = oc & 1;
                const int oo  = oc >> 1;
                const float* src = gw + (size_t)oo * WG_O + (size_t)(2*c2) * WG_C + wk0;
                float* dst = wl + oo * WL_O + c2 * WL_C2 + 2 * wk0;
#pragma unroll
                for (int u = 0; u < 4; ++u) {      // wk = wk0 + 144*u
                    v2f pr;
                    pr.x = src[144*u];             // cp=0 : c = c0 + 2*c2
                    pr.y = src[WG_C + 144*u];      // cp=1 : c = c0 + 2*c2 + 1
                    *(v2f*)(dst + 288*u) = pr;     // dwords (wk*2, wk*2+1)
                }
            }
        }

        // ---- stage x rows h-1..h+1: thread = one (c2,r,b), 4 w (stride 16), both cp ----
        {
#pragma unroll
            for (int i = 0; i < 3; ++i) {
                const int g  = tid + i * 256;      // [0,768)
                const int w0 = g & 15;             // consecutive across lanes
                const int b  = (g >> 4) & 7;
                const int r  = (g >> 7) % 3;
                const int c2 = g / 384;            // 0..1
                const int row = h - 1 + r;
                const bool ok = (row >= 0) && (row < H_);
                const float* src = x + (size_t)b * (C_*H_*W_)
                                     + (size_t)(c0 + 2*c2) * (H_*W_)
                                     + (size_t)row * W_ + w0;
                float* dst = xl + c2 * XL_C2 + r * XL_R + b * XL_B + (w0 + 1) * 2;
#pragma unroll
                for (int u = 0; u < 4; ++u) {      // w = w0 + 16*u
                    v2f pr; pr.x = 0.f; pr.y = 0.f;
                    if (ok) { pr.x = src[16*u]; pr.y = src[H_*W_ + 16*u]; }
                    *(v2f*)(dst + 32*u) = pr;
                }
            }
        }

        __syncthreads();

        // ---- compute: per pixel, 9 k-steps of V_WMMA_F32_16X16X4_F32 ----
        // K-chunk index = tap*4 + c_local ; step j covers tap = j, c_local = 0..3
        // Each operand = one ds_load_b64 (cp pair), base + compile-time immediate.
#pragma unroll
        for (int p = 0; p < 8; ++p) {
#pragma unroll
            for (int j = 0; j < KT_; ++j) {
                const int dh = j / 3, dw = j % 3;
                v2f A  = *(const v2f*)(xl + aB + dh * XL_R + (p + dw) * 2);
                v2f Bm = *(const v2f*)(wl + bB + (p * KT_ + j) * 2);
                acc[p] = __builtin_amdgcn_wmma_f32_16x16x4_f32(
                    /*neg_a=*/false, A, /*neg_b=*/false, Bm,
                    /*c_mod=*/(short)0, acc[p],
                    /*reuse_a=*/false, /*reuse_b=*/false);
            }
        }
    }

    // ---- epilogue: D VGPR r -> (M=r+8*half, N=l15); valid rows are b=r, lanes 0..15
    if (lane < 16) {
#pragma unroll
        for (int p = 0; p < 8; ++p) {
            const int w = wbase + p;
            float* ob = out + (size_t)(o0 + l15) * (H_*W_) + (size_t)h * W_ + w;
#pragma unroll
            for (int r = 0; r < 8; ++r) {        // r == batch index
                ob[(size_t)r * (O_*H_*W_)] = acc[p][r];
            }
        }
    }
}

extern "C" void kernel_launch(void* const* d_in, const int* in_sizes, int n_in,
                              void* d_out, int out_size, void* d_ws, size_t ws_size,
                              hipStream_t stream) {
    (void)in_sizes; (void)n_in; (void)out_size; (void)d_ws; (void)ws_size;
    const float* x = (const float*)d_in[0];
    const float* w = (const float*)d_in[1];
    float* out = (float*)d_out;

    const size_t shmem = (size_t)LDS_FLOATS * sizeof(float);   // ~194 KB
    hipFuncSetAttribute((const void*)lc2d_wmma_f32,
                        hipFuncAttributeMaxDynamicSharedMemorySize, (int)shmem);

    dim3 grid((O_ / OT) * H_);   // 4 o-tiles * 64 rows = 256 blocks
    dim3 block(256);             // 8 waves; wave i -> pixels w in [8i, 8i+8)
    hipLaunchKernelGGL(lc2d_wmma_f32, grid, block, shmem, stream, x, w, out);
}